// EncoLoss_32152125177945
// MI455X (gfx1250) — compile-verified
//
#include <hip/hip_runtime.h>
#include <stdint.h>

// Problem constants (from reference): B=64, N=32768, K=128, T=TMAX=512
#define B_    64
#define N_    32768
#define K_    128
#define T_    512
#define TMAX_ 512

#define AS3 __attribute__((address_space(3)))

// 32-bit LDS byte offset of a generic pointer that points into LDS.
__device__ __forceinline__ unsigned lds_addr32(const void* p) {
    return (unsigned)(unsigned long long)(AS3 const char*)p;
}

// CDNA5 async global->LDS copy, 16 bytes per lane. Tracked with ASYNCcnt.
__device__ __forceinline__ void async_load_b128(const void* gsrc, void* ldst) {
    unsigned            loff = lds_addr32(ldst);
    unsigned long long  ga   = (unsigned long long)gsrc;
    asm volatile("global_load_async_to_lds_b128 %0, %1, off"
                 :: "v"(loff), "v"(ga)
                 : "memory");
}

__device__ __forceinline__ void wait_asynccnt0() {
    asm volatile("s_wait_asynccnt 0x0" ::: "memory");
}

// ---------------------------------------------------------------------------
// Kernel 0: zero the scatter table (reference semantics: zeros + mode='drop';
// workspace is poisoned with 0xAA, so we must clear it ourselves).
// ---------------------------------------------------------------------------
__global__ void zero_table_kernel(float4* __restrict__ t, int n4) {
    int i = blockIdx.x * blockDim.x + threadIdx.x;
    const int stride = gridDim.x * blockDim.x;
    float4 z; z.x = 0.f; z.y = 0.f; z.z = 0.f; z.w = 0.f;
    #pragma unroll
    for (int k = 0; k < 4; ++k) {
        int j = i + k * stride;
        if (j < n4) t[j] = z;
    }
}

// ---------------------------------------------------------------------------
// Kernel 1: build table[b,k,tval] = traj_xy[b,k,t], tval = round(s4*10).
// One block per (b,k): 512 * 5 floats = 10240 B, async-staged into LDS
// (640 b128 chunks, 5 per lane across 128 lanes), then scattered as float2.
// ---------------------------------------------------------------------------
__global__ void build_table_kernel(const float* __restrict__ traj,
                                   float2* __restrict__ table) {
    __shared__ float lds[T_ * 5];
    const int    tid = threadIdx.x;          // 128 threads = 4 waves
    const size_t bk  = blockIdx.x;           // 0 .. B*K-1
    const float* src = traj + bk * (size_t)(T_ * 5);

    #pragma unroll
    for (int c = 0; c < 5; ++c) {
        const int chunk = tid + c * 128;     // 16-byte chunk index
        async_load_b128(src + chunk * 4, &lds[chunk * 4]);
    }
    wait_asynccnt0();      // this wave's async loads have landed in LDS
    __syncthreads();       // all waves' loads have landed

    float2* dst = table + bk * (size_t)TMAX_;
    #pragma unroll
    for (int r = 0; r < 4; ++r) {
        const int t  = tid + r * 128;
        const float x  = lds[t * 5 + 0];
        const float y  = lds[t * 5 + 1];
        const int   tv = __float2int_rn(lds[t * 5 + 4] * 10.0f);
        if (tv >= 0 && tv < TMAX_) {         // mode='drop'
            float2 v; v.x = x; v.y = y;
            dst[tv] = v;
        }
    }
}

// ---------------------------------------------------------------------------
// Kernel 2: per-block partial (sum of masked L1 diffs, count of valid).
// 1024 blocks x 256 threads x 8 elements = 2M elements. Table gathers hit L2
// (32 MB table << 192 MB L2). Deterministic: fixed reduction tree, no atomics.
// ---------------------------------------------------------------------------
__global__ void loss_partial_kernel(const float*  __restrict__ state,
                                    const int*    __restrict__ time_idx,
                                    const int*    __restrict__ uids,
                                    const float2* __restrict__ table,
                                    float2*       __restrict__ partials) {
    const int    tid  = threadIdx.x;                 // 256 threads = 8 waves
    const size_t base = (size_t)blockIdx.x * 2048;
    float s = 0.f, c = 0.f;

    #pragma unroll
    for (int k = 0; k < 8; ++k) {
        const size_t i   = base + (size_t)k * 256 + tid;
        const int    uid = uids[i];
        if (uid >= 0) {
            int t = time_idx[i];
            t = t < 0 ? 0 : (t > TMAX_ - 1 ? TMAX_ - 1 : t);
            const int    obj = uid > K_ - 1 ? K_ - 1 : uid;
            const size_t b   = i >> 15;              // N = 2^15
            const float2 tg  = table[(b * K_ + obj) * (size_t)TMAX_ + t];
            const float2 xy  = *(const float2*)(state + i * 4);
            s += fabsf(xy.x - tg.x) + fabsf(xy.y - tg.y);
            c += 1.0f;
        }
    }

    // wave32 reduction
    #pragma unroll
    for (int off = 16; off > 0; off >>= 1) {
        s += __shfl_down(s, off, 32);
        c += __shfl_down(c, off, 32);
    }
    __shared__ float2 wpart[8];
    if ((tid & 31) == 0) { float2 v; v.x = s; v.y = c; wpart[tid >> 5] = v; }
    __syncthreads();
    if (tid == 0) {
        float S = 0.f, C = 0.f;
        #pragma unroll
        for (int w = 0; w < 8; ++w) { S += wpart[w].x; C += wpart[w].y; }
        float2 v; v.x = S; v.y = C;
        partials[blockIdx.x] = v;
    }
}

// ---------------------------------------------------------------------------
// Kernel 3: deterministic final reduce of 1024 partials; loss = S / C.
// ---------------------------------------------------------------------------
__global__ void final_reduce_kernel(const float2* __restrict__ partials,
                                    float* __restrict__ out) {
    const int tid = threadIdx.x;                     // 256 threads
    float s = 0.f, c = 0.f;
    #pragma unroll
    for (int j = 0; j < 4; ++j) {
        const float2 p = partials[tid + j * 256];
        s += p.x; c += p.y;
    }
    #pragma unroll
    for (int off = 16; off > 0; off >>= 1) {
        s += __shfl_down(s, off, 32);
        c += __shfl_down(c, off, 32);
    }
    __shared__ float2 wpart[8];
    if ((tid & 31) == 0) { float2 v; v.x = s; v.y = c; wpart[tid >> 5] = v; }
    __syncthreads();
    if (tid == 0) {
        float S = 0.f, C = 0.f;
        #pragma unroll
        for (int w = 0; w < 8; ++w) { S += wpart[w].x; C += wpart[w].y; }
        out[0] = S / C;
    }
}

// ---------------------------------------------------------------------------
extern "C" void kernel_launch(void* const* d_in, const int* in_sizes, int n_in,
                              void* d_out, int out_size, void* d_ws, size_t ws_size,
                              hipStream_t stream) {
    (void)in_sizes; (void)n_in; (void)out_size; (void)ws_size;

    const float* state    = (const float*)d_in[0];   // (B,N,4) f32
    const int*   time_idx = (const int*)  d_in[1];   // (B,N)   i32
    const int*   uids     = (const int*)  d_in[2];   // (B,N)   i32
    const float* traj     = (const float*)d_in[3];   // (B,K,T,5) f32
    float*       out      = (float*)d_out;

    const size_t table_bytes = (size_t)B_ * K_ * TMAX_ * sizeof(float2); // 32 MB
    float2* table    = (float2*)d_ws;
    float2* partials = (float2*)((char*)d_ws + table_bytes);

    const int n4 = (int)(table_bytes / sizeof(float4));          // 2,097,152
    zero_table_kernel <<<2048, 256, 0, stream>>>((float4*)d_ws, n4);
    build_table_kernel<<<B_ * K_, 128, 0, stream>>>(traj, table);
    loss_partial_kernel<<<1024, 256, 0, stream>>>(state, time_idx, uids, table, partials);
    final_reduce_kernel<<<1, 256, 0, stream>>>(partials, out);
}